// Model_5686536700447
// MI455X (gfx1250) — compile-verified
//
#include <hip/hip_runtime.h>

#define N_NODES 50000
#define E_EDGES 800000
#define D 128
#define K_ORD 10
#define NUM_G 64
#define LDS_STRIDE 132   // 128 + 4 pad: bank = (4*row + col) % 64, conflict-free b64 reads

typedef __attribute__((ext_vector_type(2))) float v2f;
typedef __attribute__((ext_vector_type(8))) float v8f;

// ---------------- utility ----------------
__global__ void fill_zero_kernel(float* __restrict__ p, int n) {
    int i = blockIdx.x * blockDim.x + threadIdx.x;
    if (i < n) p[i] = 0.0f;
}

// ---------------- graph norm ----------------
__global__ void degree_kernel(const int* __restrict__ src, float* __restrict__ deg) {
    int e = blockIdx.x * blockDim.x + threadIdx.x;
    if (e < E_EDGES) atomicAdd(&deg[src[e]], 1.0f);
}

__global__ void dinv_kernel(float* __restrict__ deg) {
    int i = blockIdx.x * blockDim.x + threadIdx.x;
    if (i < N_NODES) {
        float d = deg[i];
        deg[i] = (d > 0.0f) ? rsqrtf(fmaxf(d, 1.0f)) : 0.0f;
    }
}

__global__ void norm_kernel(const int* __restrict__ src, const int* __restrict__ dst,
                            const float* __restrict__ dinv, float* __restrict__ nrm) {
    int e = blockIdx.x * blockDim.x + threadIdx.x;
    if (e < E_EDGES) nrm[e] = -dinv[src[e]] * dinv[dst[e]];
}

// ---------------- Chebyshev coefficients (tiny, single thread) ----------------
__global__ void coe_kernel(const float* __restrict__ temp, float* __restrict__ coe) {
    if (threadIdx.x != 0) return;
    float t[K_ORD + 1], xj[K_ORD + 1], Tm2[K_ORD + 1], Tm1[K_ORD + 1];
    const float PI = 3.14159265358979323846f;
    for (int j = 0; j <= K_ORD; ++j) {
        t[j]  = fmaxf(temp[j], 0.0f);
        xj[j] = cosf((K_ORD - j + 0.5f) * PI / (float)(K_ORD + 1));
    }
    for (int i = 0; i <= K_ORD; ++i) {
        float s = 0.0f;
        for (int j = 0; j <= K_ORD; ++j) {
            float Ti;
            if (i == 0)      Ti = 1.0f;
            else if (i == 1) Ti = xj[j];
            else             Ti = 2.0f * xj[j] * Tm1[j] - Tm2[j];
            s += Ti * t[j];
            Tm2[j] = Tm1[j];
            Tm1[j] = Ti;
        }
        coe[i] = (2.0f / (float)(K_ORD + 1)) * s;
    }
}

// ---------------- sparse propagate: out[dst] += norm * x[src] ----------------
// One thread handles 4 features of one edge (32 threads per edge row of 128).
__global__ void propagate_kernel(const float* __restrict__ x, const int* __restrict__ src,
                                 const int* __restrict__ dst, const float* __restrict__ nrm,
                                 float* __restrict__ out) {
    int gid = blockIdx.x * blockDim.x + threadIdx.x;
    if (gid >= E_EDGES * 32) return;
    int e = gid >> 5;
    int f = (gid & 31) << 2;
    float w = nrm[e];
    const float4 v = *reinterpret_cast<const float4*>(x + src[e] * D + f);
    float* o = out + dst[e] * D + f;
    atomicAdd(o + 0, w * v.x);
    atomicAdd(o + 1, w * v.y);
    atomicAdd(o + 2, w * v.z);
    atomicAdd(o + 3, w * v.w);
}

// ---------------- acc = coe0/2 * x + coe1 * tx1 ----------------
__global__ void cheb_init_kernel(const float* __restrict__ x, const float* __restrict__ tx1,
                                 const float* __restrict__ coe, float* __restrict__ acc) {
    int i = blockIdx.x * blockDim.x + threadIdx.x;
    if (i >= N_NODES * (D / 4)) return;
    float c0 = 0.5f * coe[0], c1 = coe[1];
    float4 a = reinterpret_cast<const float4*>(x)[i];
    float4 b = reinterpret_cast<const float4*>(tx1)[i];
    float4 r;
    r.x = c0 * a.x + c1 * b.x; r.y = c0 * a.y + c1 * b.y;
    r.z = c0 * a.z + c1 * b.z; r.w = c0 * a.w + c1 * b.w;
    reinterpret_cast<float4*>(acc)[i] = r;
}

// ---------------- tx2 = 2*P(tx1) - tx0 ; acc += coe[i]*tx2 ----------------
__global__ void cheb_step_kernel(float* __restrict__ tx2, const float* __restrict__ tx0,
                                 float* __restrict__ acc, const float* __restrict__ coe, int ci) {
    int i = blockIdx.x * blockDim.x + threadIdx.x;
    if (i >= N_NODES * (D / 4)) return;
    float c = coe[ci];
    float4 t2 = reinterpret_cast<float4*>(tx2)[i];
    float4 t0 = reinterpret_cast<const float4*>(tx0)[i];
    float4 a  = reinterpret_cast<float4*>(acc)[i];
    t2.x = 2.0f * t2.x - t0.x; t2.y = 2.0f * t2.y - t0.y;
    t2.z = 2.0f * t2.z - t0.z; t2.w = 2.0f * t2.w - t0.w;
    reinterpret_cast<float4*>(tx2)[i] = t2;
    a.x += c * t2.x; a.y += c * t2.y; a.z += c * t2.z; a.w += c * t2.w;
    reinterpret_cast<float4*>(acc)[i] = a;
}

// ---------------- WMMA f32 GEMM: out[M,128] = act(A[M,128] @ W[128,128]^T + b) ----------------
// grid.x = M/16 row tiles; 8 waves per block, wave w owns output columns [16w, 16w+16).
// A tile (16x128, 8KB) is staged once per block into LDS (padded stride 132) and
// shared by all 8 waves instead of 8x redundant global reads.
// V_WMMA_F32_16X16X4_F32 layouts (ISA 7.12.2):
//   A 16x4 : lane half h selects K-pair, VGPR v selects within pair -> K = 2h + v, M = lane&15
//   B 4x16 : same K mapping, N = lane&15   (B[k][n] = W[n][k], i.e. W row n)
//   C/D    : M = v + 8h, N = lane&15
__global__ void __launch_bounds__(256) gemm_bias_act_kernel(
        const float* __restrict__ A, const float* __restrict__ W,
        const float* __restrict__ bias, float* __restrict__ out, int relu) {
    __shared__ float As[16 * LDS_STRIDE];

    const int tid  = threadIdx.x;
    const int lane = tid & 31;
    const int wave = tid >> 5;
    const int h    = lane >> 4;
    const int lm   = lane & 15;
    const int m0   = blockIdx.x * 16;
    const int n0   = wave * 16;

    // cooperative A-tile stage: 2048 floats, 8 per thread (two float4 stores)
    {
        int r = tid >> 4;         // 0..15
        int c = (tid & 15) * 8;   // 0..120
        const float* s = A + (m0 + r) * D + c;
        float4 p0 = *reinterpret_cast<const float4*>(s);
        float4 p1 = *reinterpret_cast<const float4*>(s + 4);
        *reinterpret_cast<float4*>(&As[r * LDS_STRIDE + c])     = p0;
        *reinterpret_cast<float4*>(&As[r * LDS_STRIDE + c + 4]) = p1;
    }
    __syncthreads();

    const float* arow = &As[lm * LDS_STRIDE];
    const float* wrow = W + (n0 + lm) * D;

    v8f c = {0.f, 0.f, 0.f, 0.f, 0.f, 0.f, 0.f, 0.f};
#pragma unroll
    for (int k = 0; k < D; k += 4) {
        v2f a = *reinterpret_cast<const v2f*>(arow + k + 2 * h);
        v2f b = *reinterpret_cast<const v2f*>(wrow + k + 2 * h);
        c = __builtin_amdgcn_wmma_f32_16x16x4_f32(
                /*neg_a=*/false, a, /*neg_b=*/false, b,
                /*c_mod=*/(short)0, c, /*reuse_a=*/false, /*reuse_b=*/false);
    }

    const int n = n0 + lm;
    const float bn = bias[n];
#pragma unroll
    for (int v = 0; v < 8; ++v) {
        int m = m0 + v + 8 * h;
        float val = c[v] + bn;
        if (relu) val = fmaxf(val, 0.0f);
        out[m * D + n] = val;
    }
}

// ---------------- mean pool ----------------
__global__ void pool_kernel(const float* __restrict__ hbuf, const int* __restrict__ batch,
                            float* __restrict__ gsum, float* __restrict__ cnt) {
    int gid = blockIdx.x * blockDim.x + threadIdx.x;
    if (gid >= N_NODES * 32) return;
    int node = gid >> 5;
    int f = (gid & 31) << 2;
    int b = batch[node];
    const float4 v = *reinterpret_cast<const float4*>(hbuf + node * D + f);
    float* o = gsum + b * D + f;
    atomicAdd(o + 0, v.x); atomicAdd(o + 1, v.y);
    atomicAdd(o + 2, v.z); atomicAdd(o + 3, v.w);
    if ((gid & 31) == 0) atomicAdd(&cnt[b], 1.0f);
}

__global__ void mean_kernel(float* __restrict__ g, const float* __restrict__ cnt) {
    int i = blockIdx.x * blockDim.x + threadIdx.x;
    if (i < NUM_G * D) g[i] /= fmaxf(cnt[i >> 7], 1.0f);
}

// ---------------- host ----------------
static inline unsigned nblk(long n) { return (unsigned)((n + 255) / 256); }

extern "C" void kernel_launch(void* const* d_in, const int* in_sizes, int n_in,
                              void* d_out, int out_size, void* d_ws, size_t ws_size,
                              hipStream_t stream) {
    const float* x1      = (const float*)d_in[0];
    const int*   edge1   = (const int*)  d_in[1];
    const float* x2      = (const float*)d_in[2];
    const int*   edge2   = (const int*)  d_in[3];
    const int*   batch   = (const int*)  d_in[4];
    const float* temp    = (const float*)d_in[5];
    const float* lin1_w  = (const float*)d_in[6];
    const float* lin1_b  = (const float*)d_in[7];
    const float* proj_w1 = (const float*)d_in[8];
    const float* proj_b1 = (const float*)d_in[9];
    const float* proj_w2 = (const float*)d_in[10];
    const float* proj_b2 = (const float*)d_in[11];
    float* zout = (float*)d_out;

    // workspace layout (floats)
    float* ws   = (float*)d_ws;
    float* deg  = ws;                       // N
    float* nrm  = deg + N_NODES;            // E
    float* coe  = nrm + E_EDGES;            // 16
    float* bufA = coe + 16;                 // N*D
    float* bufB = bufA + N_NODES * D;       // N*D
    float* bufC = bufB + N_NODES * D;       // N*D
    float* accb = bufC + N_NODES * D;       // N*D
    float* g1   = accb + N_NODES * D;       // 64*128
    float* g2   = g1 + NUM_G * D;           // 64*128
    float* cnt  = g2 + NUM_G * D;           // 64
    float* hid  = cnt + 64;                 // 64*128

    const float* xs[2]    = {x1, x2};
    const int*   edges[2] = {edge1, edge2};
    float*       gs[2]    = {g1, g2};
    float*       bufs[3]  = {bufA, bufB, bufC};

    coe_kernel<<<1, 32, 0, stream>>>(temp, coe);

    for (int view = 0; view < 2; ++view) {
        const float* x   = xs[view];
        const int*   src = edges[view];
        const int*   dst = edges[view] + E_EDGES;

        fill_zero_kernel<<<nblk(N_NODES), 256, 0, stream>>>(deg, N_NODES);
        degree_kernel<<<nblk(E_EDGES), 256, 0, stream>>>(src, deg);
        dinv_kernel<<<nblk(N_NODES), 256, 0, stream>>>(deg);
        norm_kernel<<<nblk(E_EDGES), 256, 0, stream>>>(src, dst, deg, nrm);

        // tx1 = P(x); acc = coe0/2 * x + coe1 * tx1
        fill_zero_kernel<<<nblk((long)N_NODES * D), 256, 0, stream>>>(bufA, N_NODES * D);
        propagate_kernel<<<nblk((long)E_EDGES * 32), 256, 0, stream>>>(x, src, dst, nrm, bufA);
        cheb_init_kernel<<<nblk((long)N_NODES * 32), 256, 0, stream>>>(x, bufA, coe, accb);

        const float* t0 = x;
        const float* t1 = bufA;
        for (int i = 2; i <= K_ORD; ++i) {
            float* t2 = bufs[(i - 1) % 3];   // cycles B,C,A,... always distinct from t0,t1
            fill_zero_kernel<<<nblk((long)N_NODES * D), 256, 0, stream>>>(t2, N_NODES * D);
            propagate_kernel<<<nblk((long)E_EDGES * 32), 256, 0, stream>>>(t1, src, dst, nrm, t2);
            cheb_step_kernel<<<nblk((long)N_NODES * 32), 256, 0, stream>>>(t2, t0, accb, coe, i);
            t0 = t1; t1 = t2;
        }

        // h = relu(acc @ lin1_w^T + lin1_b)  — WMMA GEMM, 3125 row tiles
        float* h = bufs[K_ORD % 3];          // free buffer after recurrence
        gemm_bias_act_kernel<<<N_NODES / 16, 256, 0, stream>>>(accb, lin1_w, lin1_b, h, 1);

        fill_zero_kernel<<<nblk(NUM_G * D), 256, 0, stream>>>(gs[view], NUM_G * D);
        fill_zero_kernel<<<1, 256, 0, stream>>>(cnt, NUM_G);
        pool_kernel<<<nblk((long)N_NODES * 32), 256, 0, stream>>>(h, batch, gs[view], cnt);
        mean_kernel<<<nblk(NUM_G * D), 256, 0, stream>>>(gs[view], cnt);
    }

    // projection head (WMMA GEMMs on [64,128])
    for (int view = 0; view < 2; ++view) {
        gemm_bias_act_kernel<<<NUM_G / 16, 256, 0, stream>>>(gs[view], proj_w1, proj_b1, hid, 1);
        gemm_bias_act_kernel<<<NUM_G / 16, 256, 0, stream>>>(hid, proj_w2, proj_b2,
                                                             zout + view * NUM_G * D, 0);
    }
}